// SAM_50440095924353
// MI455X (gfx1250) — compile-verified
//
#include <hip/hip_runtime.h>

typedef __attribute__((ext_vector_type(2))) float v2f;
typedef __attribute__((ext_vector_type(8))) float v8f;

#define NPIX 256            // 16*16 elements per (b,c) slab
#define ROWS_PER_WAVE 16    // one 16-row WMMA tile per wave
#define WAVES_PER_BLOCK 8

// out[p] = dot(x[p, 0:256], G) + bias   for p in [0, 131072)
// computed as chained V_WMMA_F32_16X16X4_F32: A = 16 rows x 4 K, B = G-chunk
// broadcast into all 16 columns, D accumulates the dot in every column.
__global__ __launch_bounds__(256) void pool_linear_wmma(
    const float* __restrict__ x, const float* __restrict__ Wv,
    const float* __restrict__ bias, float* __restrict__ out)
{
    __shared__ __align__(16) float Gs[NPIX];
    const int t = threadIdx.x;

    // Build the effective 16x16 weight map G from the shared Linear(36,1)
    // weights: G[h,w] = (1/25) * sum of W over 5x5/s2 windows covering (h,w).
    {
        const int h = t >> 4, w = t & 15;
        float acc = 0.f;
        #pragma unroll
        for (int oh = 0; oh < 6; ++oh) {
            const int dh = h - 2 * oh;
            #pragma unroll
            for (int ow = 0; ow < 6; ++ow) {
                const int dw = w - 2 * ow;
                if (dh >= 0 && dh < 5 && dw >= 0 && dw < 5)
                    acc += Wv[oh * 6 + ow];
            }
        }
        Gs[t] = acc * 0.04f;   // 1/25
    }
    __syncthreads();

    const int lane = t & 31;
    const int wave = t >> 5;
    const int hi   = lane >> 4;    // 0 -> K pair {0,1}; 1 -> K pair {2,3}
    const int mrow = lane & 15;    // A-matrix row M
    const long tile = (long)blockIdx.x * WAVES_PER_BLOCK + wave;
    const float* __restrict__ xrow = x + (tile * ROWS_PER_WAVE + mrow) * (long)NPIX;

    // Two accumulators to pipeline the WMMA dependency chain.
    v8f acc0 = {};
    v8f acc1 = {};

    #pragma unroll 4
    for (int kk = 0; kk < NPIX; kk += 8) {
        const int k0 = kk + 2 * hi;
        v2f a0 = *(const v2f*)(xrow + k0);          // A: x[row, k0], x[row, k0+1]
        v2f b0 = *(const v2f*)(&Gs[k0]);            // B: G[k0], G[k0+1] (bcast over N)
        acc0 = __builtin_amdgcn_wmma_f32_16x16x4_f32(
                   false, a0, false, b0, (short)0, acc0, false, false);

        const int k1 = kk + 4 + 2 * hi;
        v2f a1 = *(const v2f*)(xrow + k1);
        v2f b1 = *(const v2f*)(&Gs[k1]);
        acc1 = __builtin_amdgcn_wmma_f32_16x16x4_f32(
                   false, a1, false, b1, (short)0, acc1, false, false);
    }

    const v8f acc = acc0 + acc1;
    const float bb = bias[0];

    // D layout: VGPR r holds row r (lanes 0-15) / row 8+r (lanes 16-31),
    // every column equals the dot product -> lanes 0 and 16 store 8 rows each.
    if (mrow == 0) {
        const long obase = tile * ROWS_PER_WAVE + (long)hi * 8;
        #pragma unroll
        for (int r = 0; r < 8; ++r)
            out[obase + r] = acc[r] + bb;
    }
}

extern "C" void kernel_launch(void* const* d_in, const int* in_sizes, int n_in,
                              void* d_out, int out_size, void* d_ws, size_t ws_size,
                              hipStream_t stream) {
    (void)in_sizes; (void)n_in; (void)out_size; (void)d_ws; (void)ws_size;
    const float* x  = (const float*)d_in[0];   // [256, 512, 16, 16] fp32
    const float* W  = (const float*)d_in[1];   // [1, 36] fp32
    const float* b  = (const float*)d_in[2];   // [1] fp32
    float* out = (float*)d_out;                // [256, 512] fp32

    // 131072 rows / (8 waves * 16 rows per block) = 1024 blocks
    dim3 grid(1024), block(256);
    hipLaunchKernelGGL(pool_linear_wmma, grid, block, 0, stream, x, W, b, out);
}